// ConstraintLayer_27754078666768
// MI455X (gfx1250) — compile-verified
//
#include <hip/hip_runtime.h>

// ============================================================================
// ConstraintLayer: x = y - A^T (A A^T)^{-1} (A y - b)   (projection onto Ax=b)
//   y: [B,64] f32, A: [16,64] f32, b: [B,16] f32 -> out: [B,64] f32
//
// Roofline: 576 B/row HBM traffic (604 MB total) -> ~26us @ 23.3 TB/s.
// Factored form needs only 4.6 KFLOP/row (~185 TF fp32 to stay mem-bound),
// well within V_WMMA_F32_16X16X4_F32 throughput, at full fp32 precision.
// ============================================================================

#define NW 8            // waves per block (256 threads, wave32)
#define YSTR 68         // LDS row stride (floats) for y tile (conflict-free)
#define SSTR 18         // LDS row stride (floats) for b / mu tiles
#define WAVE_LDS (16*YSTR + 16*SSTR + 16*SSTR)   // 1664 floats = 6656 B

typedef __attribute__((ext_vector_type(2))) float v2f;
typedef __attribute__((ext_vector_type(8))) float v8f;
typedef __attribute__((__vector_size__(2 * sizeof(int)))) int i2v;  // b64 payload

// ---- fp32 WMMA (16x16x4): D = A(16x4) x B(4x16) + C ------------------------
__device__ __forceinline__ v8f wmma4(v2f a, v2f b, v8f c) {
#ifdef __HIP_DEVICE_COMPILE__
  // 8-arg pattern: (neg_a, A, neg_b, B, c_mod, C, reuse_a, reuse_b)
  return __builtin_amdgcn_wmma_f32_16x16x4_f32(false, a, false, b, (short)0, c,
                                               false, false);
#else
  (void)a; (void)b; return c;
#endif
}

#ifdef __HIP_DEVICE_COMPILE__
#define WAVE_BARRIER() __builtin_amdgcn_wave_barrier()
#else
#define WAVE_BARRIER()
#endif

#if defined(__HIP_DEVICE_COMPILE__) && \
    __has_builtin(__builtin_amdgcn_global_load_async_to_lds_b64)
#define USE_ASYNC 1
typedef __attribute__((address_space(3))) i2v* las3_t;   // __shared__ int2*
#endif

__device__ __forceinline__ void wait_async0() {
#ifdef __HIP_DEVICE_COMPILE__
#if __has_builtin(__builtin_amdgcn_s_wait_asynccnt)
  __builtin_amdgcn_s_wait_asynccnt(0);
#else
  asm volatile("s_wait_asynccnt 0x0" ::: "memory");
#endif
#endif
}

// ============================================================================
// Prep (one block): G = A A^T; Ginv via double Gauss-Jordan (SPD, no pivot);
// write constants to workspace:
//   cst[0..1023]    : Q = Ginv @ A          (16x64 row-major)
//   cst[1024..1279] : -Ginv                 (16x16 row-major, symmetric)
//   cst[1280..2303] : -A^T                  (64x16 row-major)
// ============================================================================
__global__ __launch_bounds__(256) void constraint_prep(
    const float* __restrict__ A, float* __restrict__ cst) {
  __shared__ float sA[16 * 64];
  __shared__ double M[16][32];
  const int t = threadIdx.x;

  for (int e = t; e < 1024; e += 256) sA[e] = A[e];
  __syncthreads();

  {
    const int i = t >> 4, j = t & 15;
    double s = 0.0;
    for (int k = 0; k < 64; ++k)
      s += (double)sA[i * 64 + k] * (double)sA[j * 64 + k];
    M[i][j] = s;
    M[i][16 + j] = (i == j) ? 1.0 : 0.0;
  }
  __syncthreads();

  if (t == 0) {  // serial 16x16 Gauss-Jordan in double (one-time cost)
    for (int c = 0; c < 16; ++c) {
      const double inv = 1.0 / M[c][c];
      for (int j = 0; j < 32; ++j) M[c][j] *= inv;
      for (int r = 0; r < 16; ++r) {
        if (r == c) continue;
        const double f = M[r][c];
        for (int j = 0; j < 32; ++j) M[r][j] -= f * M[c][j];
      }
    }
  }
  __syncthreads();

  // Q = Ginv @ A
  for (int e = t; e < 1024; e += 256) {
    const int i = e >> 6, k = e & 63;
    double s = 0.0;
    for (int j = 0; j < 16; ++j) s += M[i][16 + j] * (double)sA[j * 64 + k];
    cst[e] = (float)s;
  }
  // -Ginv
  cst[1024 + t] = -(float)M[t >> 4][16 + (t & 15)];
  // -A^T : negAT[k][i] = -A[i][k]
  for (int e = t; e < 1024; e += 256) {
    const int k = e >> 4, i = e & 15;
    cst[1280 + e] = -sA[i * 64 + k];
  }
}

// ============================================================================
// Main: persistent waves; each wave processes 16-row batch tiles.
//   stage1: mu(16x16)  = y(16x64) @ Q^T  +  b(16x16) @ (-Ginv)     [20 WMMA]
//   transpose mu D-layout -> A-operand via wave-private LDS
//   stage2: out(16x64) = y  +  mu @ (-A)                           [16 WMMA]
// ============================================================================
__global__ __launch_bounds__(256) void constraint_main(
    const float* __restrict__ y, const float* __restrict__ b,
    const float* __restrict__ cst, float* __restrict__ out, int ntiles) {
  __shared__ float smem[NW * WAVE_LDS];

  const int lane = threadIdx.x & 31;
  const int wave = threadIdx.x >> 5;
  const int half = lane >> 4;   // which 16-lane half
  const int l16  = lane & 15;

  float* ylds = smem + wave * WAVE_LDS;    // 16 x YSTR
  float* blds = ylds + 16 * YSTR;          // 16 x SSTR
  float* mlds = blds + 16 * SSTR;          // 16 x SSTR

  const float* Q   = cst;           // 16x64
  const float* nG  = cst + 1024;    // 16x16  (-Ginv, symmetric)
  const float* nAT = cst + 1280;    // 64x16  (-A^T)

  // ---- constant B-operands, loaded once per wave -------------------------
  // B-operand of matrix B(KxN) == A-operand layout of B^T(NxK):
  //   lane reads row-major B^T [l16][4k + 2*half + {0,1}] as float2.
  v2f qv[16], gv[4], av[4][4];
#pragma unroll
  for (int k = 0; k < 16; ++k)
    qv[k] = *(const v2f*)(Q + l16 * 64 + 4 * k + 2 * half);       // Q^T B-op
#pragma unroll
  for (int k = 0; k < 4; ++k)
    gv[k] = *(const v2f*)(nG + l16 * 16 + 4 * k + 2 * half);      // -Ginv B-op
#pragma unroll
  for (int c = 0; c < 4; ++c)
#pragma unroll
    for (int k = 0; k < 4; ++k)
      av[c][k] = *(const v2f*)(nAT + (16 * c + l16) * 16 + 4 * k + 2 * half);

  const int gwave  = blockIdx.x * NW + wave;
  const int nwaves = gridDim.x * NW;

  for (int tile = gwave; tile < ntiles; tile += nwaves) {
    const float* ysrc = y + (size_t)tile * (16 * 64);
    const float* bsrc = b + (size_t)tile * (16 * 16);

    // ---- stage y (4KB) and b (1KB) tiles into padded wave-private LDS ----
#ifdef USE_ASYNC
#pragma unroll
    for (int i = 0; i < 16; ++i)
      __builtin_amdgcn_global_load_async_to_lds_b64(
          (i2v*)(ysrc + i * 64 + lane * 2),
          (las3_t)(ylds + i * YSTR + lane * 2), 0, 0);
#pragma unroll
    for (int i = 0; i < 4; ++i)
      __builtin_amdgcn_global_load_async_to_lds_b64(
          (i2v*)(bsrc + i * 64 + lane * 2),
          (las3_t)(blds + (4 * i + (lane >> 3)) * SSTR + (lane & 7) * 2), 0, 0);
    wait_async0();
    WAVE_BARRIER();
#else
#pragma unroll
    for (int i = 0; i < 16; ++i)
      *(v2f*)(ylds + i * YSTR + lane * 2) =
          *(const v2f*)(ysrc + i * 64 + lane * 2);
#pragma unroll
    for (int i = 0; i < 4; ++i)
      *(v2f*)(blds + (4 * i + (lane >> 3)) * SSTR + (lane & 7) * 2) =
          *(const v2f*)(bsrc + i * 64 + lane * 2);
    WAVE_BARRIER();
#endif

    // ---- stage 1: mu = y @ Q^T + b @ (-Ginv), K = 64 + 16 ----------------
    v8f acc = {0.f, 0.f, 0.f, 0.f, 0.f, 0.f, 0.f, 0.f};
#pragma unroll
    for (int k = 0; k < 16; ++k) {
      v2f a = *(const v2f*)(ylds + l16 * YSTR + 4 * k + 2 * half);  // y A-op
      acc = wmma4(a, qv[k], acc);
    }
#pragma unroll
    for (int k = 0; k < 4; ++k) {
      v2f a = *(const v2f*)(blds + l16 * SSTR + 4 * k + 2 * half);  // b A-op
      acc = wmma4(a, gv[k], acc);
    }

    // ---- transpose mu: D-layout -> A-operand through LDS -----------------
    // D: VGPR v, lane: row M = v + 8*half (batch), col N = l16.
#pragma unroll
    for (int v = 0; v < 8; ++v)
      mlds[(v + 8 * half) * SSTR + l16] = acc[v];
    WAVE_BARRIER();  // DS ops are in-order within a wave

    v2f mu[4];
#pragma unroll
    for (int k = 0; k < 4; ++k)
      mu[k] = *(const v2f*)(mlds + l16 * SSTR + 4 * k + 2 * half);  // mu A-op

    // ---- stage 2: out = y + mu @ (-A), 4 N-chunks of 16 ------------------
    float* odst = out + (size_t)tile * (16 * 64);
#pragma unroll
    for (int c = 0; c < 4; ++c) {
      v8f acc2;
#pragma unroll
      for (int v = 0; v < 8; ++v)                     // C initialized with y
        acc2[v] = ylds[(v + 8 * half) * YSTR + 16 * c + l16];
#pragma unroll
      for (int k = 0; k < 4; ++k)
        acc2 = wmma4(mu[k], av[c][k], acc2);
#pragma unroll
      for (int v = 0; v < 8; ++v)
        odst[(v + 8 * half) * 64 + 16 * c + l16] = acc2[v];
    }
  }
}

// ============================================================================
extern "C" void kernel_launch(void* const* d_in, const int* in_sizes, int n_in,
                              void* d_out, int out_size, void* d_ws,
                              size_t ws_size, hipStream_t stream) {
  const float* y = (const float*)d_in[0];  // [B,64]
  const float* A = (const float*)d_in[1];  // [16,64]
  const float* b = (const float*)d_in[2];  // [B,16]
  float* out = (float*)d_out;              // [B,64]
  float* cst = (float*)d_ws;               // 2304 floats used

  const int batch  = in_sizes[0] / 64;
  const int ntiles = batch / 16;

  constraint_prep<<<1, 256, 0, stream>>>(A, cst);

  int nb = 2048;                                  // 16384 persistent waves
  const int maxnb = (ntiles + NW - 1) / NW;
  if (nb > maxnb) nb = maxnb;
  if (nb < 1) nb = 1;
  constraint_main<<<nb, 256, 0, stream>>>(y, b, cst, out, ntiles);
}